// GraphPolicyNetwork_77670188581040
// MI455X (gfx1250) — compile-verified
//
#include <hip/hip_runtime.h>
#include <hip/hip_bf16.h>

#define N_NODES   131072
#define NODES_PG  2048
#define BATCH     64
#define E_EDGES   2097152
#define HID       128
#define KDIM      4096   // 2*NODES_PG
#define NCOL      2048   // NODES_PG

#define KSLICES   8      // K split across blocks -> 512 K per slice
#define KSLICE    (KDIM / KSLICES)
#define KCHUNK    64     // K staged in LDS per iteration
#define LDSPAD    66     // floats per LDS row (pad 64 -> 66: spreads banks, keeps 8B align)

typedef __attribute__((ext_vector_type(2))) float v2f;
typedef __attribute__((ext_vector_type(8))) float v8f;

// ---------------------------------------------------------------- zero scratch
__global__ void k_zero(float* p, int n) {
    int i = blockIdx.x * blockDim.x + threadIdx.x;
    if (i < n) p[i] = 0.0f;
}

// ------------------------------------------------- edge pass 1: agg x and deg
__global__ void k_edge1(const int* __restrict__ src, const int* __restrict__ dst,
                        const float* __restrict__ x,
                        float* __restrict__ aggx, float* __restrict__ deg) {
    int e = blockIdx.x * blockDim.x + threadIdx.x;
    if (e >= E_EDGES) return;
    int s = src[e], d = dst[e];
    atomicAdd(&aggx[d], x[s]);
    atomicAdd(&deg[d], 1.0f);
}

// --------------------- per-node fused MLP: s_self, s_neigh from x and mean(x)
__global__ void k_node(const float* __restrict__ x,
                       const float* __restrict__ aggx, const float* __restrict__ deg,
                       const float* __restrict__ W1s, const float* __restrict__ W1n,
                       const float* __restrict__ b1,
                       const float* __restrict__ W2s, const float* __restrict__ W2n,
                       float* __restrict__ s_self, float* __restrict__ s_neigh) {
    __shared__ float lw1s[HID], lw1n[HID], lb1[HID], lw2s[HID], lw2n[HID];
    if (threadIdx.x < HID) {
        int h = threadIdx.x;
        lw1s[h] = W1s[h]; lw1n[h] = W1n[h]; lb1[h] = b1[h];
        lw2s[h] = W2s[h]; lw2n[h] = W2n[h];
    }
    __syncthreads();
    int v = blockIdx.x * blockDim.x + threadIdx.x;
    if (v >= N_NODES) return;
    float xv = x[v];
    float dg = deg[v];
    float m  = aggx[v] / fmaxf(dg, 1.0f);
    float ss = 0.0f, sn = 0.0f;
    #pragma unroll 8
    for (int h = 0; h < HID; ++h) {
        float t = tanhf(fmaf(xv, lw1s[h], fmaf(m, lw1n[h], lb1[h])));
        ss = fmaf(t, lw2s[h], ss);
        sn = fmaf(t, lw2n[h], sn);
    }
    s_self[v]  = ss;
    s_neigh[v] = sn;
}

// ------------------------------------------- edge pass 2: aggregate s_neigh
__global__ void k_edge2(const int* __restrict__ src, const int* __restrict__ dst,
                        const float* __restrict__ s_neigh, float* __restrict__ aggs) {
    int e = blockIdx.x * blockDim.x + threadIdx.x;
    if (e >= E_EDGES) return;
    atomicAdd(&aggs[dst[e]], s_neigh[src[e]]);
}

// -------- finish out1 = s_self + mean(s_neigh) + b2 ; build h2 = tanh(concat)
__global__ void k_finish(const float* __restrict__ x,
                         const float* __restrict__ s_self, const float* __restrict__ aggs,
                         const float* __restrict__ deg, const float* __restrict__ b2,
                         float* __restrict__ out1, float* __restrict__ h2) {
    int v = blockIdx.x * blockDim.x + threadIdx.x;
    if (v >= N_NODES) return;
    float o = s_self[v] + aggs[v] / fmaxf(deg[v], 1.0f) + b2[0];
    out1[v] = o;
    int b = v >> 11;            // / 2048
    int j = v & (NODES_PG - 1); // % 2048
    h2[b * KDIM + j]            = tanhf(o);
    h2[b * KDIM + NODES_PG + j] = tanhf(x[v]);
}

// ------------------------------------ out2 = b3 (broadcast), GEMM accumulates
__global__ void k_init_out2(const float* __restrict__ b3, float* __restrict__ out2) {
    int i = blockIdx.x * blockDim.x + threadIdx.x;
    if (i < BATCH * NCOL) out2[i] = b3[i & (NCOL - 1)];
}

// --------------------------- out2 += h2 @ W3 via V_WMMA_F32_16X16X4_F32
// h2: [64, 4096] row-major; W3: [4096, 2048] row-major; out2: [64, 2048].
// Block = 8 waves covering a 128-col super-stripe, one K-slice of 512.
// Each wave: 64x16 output (4 M-tile accumulators), B fragment loaded once and
// shared by 4 WMMAs; A staged through LDS (coalesced, bank-spread).
__global__ void __launch_bounds__(256)
k_gemm_wmma(const float* __restrict__ h2, const float* __restrict__ W3,
            float* __restrict__ out2) {
    __shared__ float ldsA[BATCH * LDSPAD];         // 64 x 66 floats = 16.5 KB

    const int nsuper = blockIdx.x & 15;            // 16 super-stripes of 128 cols
    const int kslice = blockIdx.x >> 4;            // 0..7
    const int wave   = threadIdx.x >> 5;           // 0..7
    const int lane   = threadIdx.x & 31;
    const int half   = lane >> 4;                  // 0: K=0..1, 1: K=2..3
    const int lid    = lane & 15;
    const int n0     = nsuper * 128 + wave * 16;
    const int kbase  = kslice * KSLICE;

    v8f acc0 = {}, acc1 = {}, acc2 = {}, acc3 = {};

    for (int kc = 0; kc < KSLICE; kc += KCHUNK) {
        __syncthreads();                            // previous chunk consumed
        // Cooperative, coalesced A-chunk load: 64 rows x 64 K
        #pragma unroll
        for (int i = threadIdx.x; i < BATCH * KCHUNK; i += 256) {
            int r = i >> 6;                         // row 0..63
            int c = i & (KCHUNK - 1);               // k within chunk
            ldsA[r * LDSPAD + c] = h2[(size_t)r * KDIM + kbase + kc + c];
        }
        __syncthreads();

        #pragma unroll 4
        for (int ks = 0; ks < KCHUNK; ks += 4) {
            const int kk = kbase + kc + ks;
            // B fragment: lanes 0-15 -> K rows kk,kk+1; lanes 16-31 -> kk+2,kk+3
            const float* bp = W3 + (size_t)(kk + 2 * half) * NCOL + n0 + lid;
            v2f b;
            b.x = bp[0];
            b.y = bp[NCOL];
            const int ao = ks + 2 * half;
            v2f a0, a1, a2, a3;
            a0.x = ldsA[(lid     ) * LDSPAD + ao]; a0.y = ldsA[(lid     ) * LDSPAD + ao + 1];
            a1.x = ldsA[(lid + 16) * LDSPAD + ao]; a1.y = ldsA[(lid + 16) * LDSPAD + ao + 1];
            a2.x = ldsA[(lid + 32) * LDSPAD + ao]; a2.y = ldsA[(lid + 32) * LDSPAD + ao + 1];
            a3.x = ldsA[(lid + 48) * LDSPAD + ao]; a3.y = ldsA[(lid + 48) * LDSPAD + ao + 1];
            acc0 = __builtin_amdgcn_wmma_f32_16x16x4_f32(false, a0, false, b, (short)0, acc0, false, false);
            acc1 = __builtin_amdgcn_wmma_f32_16x16x4_f32(false, a1, false, b, (short)0, acc1, false, false);
            acc2 = __builtin_amdgcn_wmma_f32_16x16x4_f32(false, a2, false, b, (short)0, acc2, false, false);
            acc3 = __builtin_amdgcn_wmma_f32_16x16x4_f32(false, a3, false, b, (short)0, acc3, false, false);
        }
    }

    // C/D layout: VGPR r -> M = r (lanes 0-15) / r+8 (lanes 16-31), N = lid
    #pragma unroll
    for (int r = 0; r < 8; ++r) {
        const int mrow = r + 8 * half;
        const size_t col = n0 + lid;
        atomicAdd(&out2[(size_t)(mrow     ) * NCOL + col], acc0[r]);
        atomicAdd(&out2[(size_t)(mrow + 16) * NCOL + col], acc1[r]);
        atomicAdd(&out2[(size_t)(mrow + 32) * NCOL + col], acc2[r]);
        atomicAdd(&out2[(size_t)(mrow + 48) * NCOL + col], acc3[r]);
    }
}

extern "C" void kernel_launch(void* const* d_in, const int* in_sizes, int n_in,
                              void* d_out, int out_size, void* d_ws, size_t ws_size,
                              hipStream_t stream) {
    const float* x   = (const float*)d_in[0];
    const int*   src = (const int*)  d_in[1];
    const int*   dst = (const int*)  d_in[2];
    const float* W1s = (const float*)d_in[3];
    const float* W1n = (const float*)d_in[4];
    const float* b1  = (const float*)d_in[5];
    const float* W2s = (const float*)d_in[6];
    const float* W2n = (const float*)d_in[7];
    const float* b2  = (const float*)d_in[8];
    const float* W3  = (const float*)d_in[9];
    const float* b3  = (const float*)d_in[10];

    float* ws      = (float*)d_ws;
    float* aggx    = ws;                     // [N]
    float* deg     = ws + 1 * N_NODES;       // [N]
    float* aggs    = ws + 2 * N_NODES;       // [N]
    float* s_self  = ws + 3 * N_NODES;       // [N]
    float* s_neigh = ws + 4 * N_NODES;       // [N]
    float* h2      = ws + 5 * N_NODES;       // [64*4096]

    float* out1 = (float*)d_out;             // [N]
    float* out2 = out1 + N_NODES;            // [64*2048]

    const int B = 256;
    // zero the three atomic accumulators (aggx, deg, aggs are contiguous)
    k_zero<<<(3 * N_NODES + B - 1) / B, B, 0, stream>>>(aggx, 3 * N_NODES);
    k_edge1<<<(E_EDGES + B - 1) / B, B, 0, stream>>>(src, dst, x, aggx, deg);
    k_node<<<(N_NODES + B - 1) / B, B, 0, stream>>>(x, aggx, deg, W1s, W1n, b1,
                                                    W2s, W2n, s_self, s_neigh);
    k_edge2<<<(E_EDGES + B - 1) / B, B, 0, stream>>>(src, dst, s_neigh, aggs);
    k_finish<<<(N_NODES + B - 1) / B, B, 0, stream>>>(x, s_self, aggs, deg, b2, out1, h2);

    k_init_out2<<<(BATCH * NCOL + B - 1) / B, B, 0, stream>>>(b3, out2);
    // 16 n-super-stripes x 8 K-slices = 128 blocks x 8 waves = 1024 waves;
    // W3 streamed exactly once, A from LDS, B fragment shared by 4 WMMAs.
    k_gemm_wmma<<<128, 256, 0, stream>>>(h2, W3, out2);
}